// GATPEEmbedding_33861522162254
// MI455X (gfx1250) — compile-verified
//
#include <hip/hip_runtime.h>
#include <math.h>

// Problem constants (from reference)
#define B_ 32
#define T_ 10
#define N_ 1000
#define C_ 96
#define E_ 8000
#define G_ (B_ * T_)     // 320 graphs
#define ET_ (E_ + N_)    // 9000 edges incl. self loops
#define NEG_SLOPE 0.2f

typedef __attribute__((ext_vector_type(2))) float v2f;
typedef __attribute__((ext_vector_type(8))) float v8f;

// ---------------------------------------------------------------------------
// float atomic max via sign-aware integer atomics (works with -inf init)
// ---------------------------------------------------------------------------
__device__ __forceinline__ void atomicMaxF(float* addr, float val) {
  if (val >= 0.0f) {
    atomicMax((int*)addr, __float_as_int(val));
  } else {
    atomicMin((unsigned int*)addr, (unsigned int)__float_as_int(val));
  }
}

// ---------------------------------------------------------------------------
// Kernel 0: init scratch + build constant tables
//   constp[t*C+c] = bl[c] + bias[c] + sinusoidal_pe(t,c)
//   tbl4[c]       = { Wl[c], Wr[c], bl[c]+br[c], att[c] }
// ---------------------------------------------------------------------------
__global__ void init_kernel(const float* __restrict__ Wl, const float* __restrict__ bl,
                            const float* __restrict__ Wr, const float* __restrict__ br,
                            const float* __restrict__ att, const float* __restrict__ bias,
                            float* __restrict__ m, float* __restrict__ den,
                            float* __restrict__ snum, float* __restrict__ constp,
                            float* __restrict__ tbl4) {
  int i = blockIdx.x * blockDim.x + threadIdx.x;
  if (i < G_ * N_) {
    m[i]    = __int_as_float((int)0xff800000);  // -inf
    den[i]  = 0.0f;
    snum[i] = 0.0f;
  }
  if (i < T_ * C_) {
    int t = i / C_;
    int c = i - t * C_;
    float d   = expf(-logf(10000.0f) * (float)(c & ~1) / (float)C_);
    float ang = (float)t * d;
    float pe  = (c & 1) ? cosf(ang) : sinf(ang);
    constp[i] = bl[c] + bias[c] + pe;
  }
  if (i < C_) {
    tbl4[4 * i + 0] = Wl[i];
    tbl4[4 * i + 1] = Wr[i];
    tbl4[4 * i + 2] = bl[i] + br[i];
    tbl4[4 * i + 3] = att[i];
  }
}

// ---------------------------------------------------------------------------
// Edge logit: e = sum_c att[c] * lrelu(x_src*Wl[c] + x_dst*Wr[c] + bl[c]+br[c])
// (FIN==1 collapses the linear layers to rank-1)
// ---------------------------------------------------------------------------
__device__ __forceinline__ float edge_logit(float xs, float xd, const float4* tbl) {
  float e = 0.0f;
#pragma unroll 8
  for (int c = 0; c < C_; ++c) {
    float4 w = tbl[c];                                 // {Wl, Wr, bl+br, att}
    float v  = fmaf(xs, w.x, fmaf(xd, w.y, w.z));
    v        = fmaxf(v, NEG_SLOPE * v);                // leaky relu (slope < 1)
    e        = fmaf(w.w, v, e);
  }
  return e;
}

__device__ __forceinline__ void decode_edge(int et, const int* __restrict__ ei,
                                            int& src, int& dst) {
  if (et < E_) {
    src = ei[et];
    dst = ei[E_ + et];
  } else {            // appended self loops
    src = et - E_;
    dst = et - E_;
  }
}

// ---------------------------------------------------------------------------
// Kernel 1: per-edge logits -> segment max over dst
// ---------------------------------------------------------------------------
__global__ void logits_max_kernel(const float* __restrict__ x, const int* __restrict__ ei,
                                  const float* __restrict__ tbl4, float* __restrict__ m) {
  __shared__ float4 s_tbl[C_];
  if (threadIdx.x < C_) s_tbl[threadIdx.x] = ((const float4*)tbl4)[threadIdx.x];
  __syncthreads();

  int idx = blockIdx.x * blockDim.x + threadIdx.x;
  if (idx >= G_ * ET_) return;
  int g  = idx / ET_;
  int et = idx - g * ET_;
  int src, dst;
  decode_edge(et, ei, src, dst);
  float xs = x[g * N_ + src];
  float xd = x[g * N_ + dst];
  float e  = edge_logit(xs, xd, s_tbl);
  atomicMaxF(&m[g * N_ + dst], e);
}

// ---------------------------------------------------------------------------
// Kernel 2: recompute logit, ex = exp(e - m[dst]);
//   den[dst]  += ex
//   snum[dst] += ex * x_src      (scalar message numerator, FIN==1)
// ---------------------------------------------------------------------------
__global__ void softmax_accum_kernel(const float* __restrict__ x, const int* __restrict__ ei,
                                     const float* __restrict__ tbl4,
                                     const float* __restrict__ m,
                                     float* __restrict__ den, float* __restrict__ snum) {
  __shared__ float4 s_tbl[C_];
  if (threadIdx.x < C_) s_tbl[threadIdx.x] = ((const float4*)tbl4)[threadIdx.x];
  __syncthreads();

  int idx = blockIdx.x * blockDim.x + threadIdx.x;
  if (idx >= G_ * ET_) return;
  int g  = idx / ET_;
  int et = idx - g * ET_;
  int src, dst;
  decode_edge(et, ei, src, dst);
  float xs = x[g * N_ + src];
  float xd = x[g * N_ + dst];
  float e  = edge_logit(xs, xd, s_tbl);
  float ex = __expf(e - m[g * N_ + dst]);
  atomicAdd(&den[g * N_ + dst], ex);
  atomicAdd(&snum[g * N_ + dst], ex * xs);
}

// ---------------------------------------------------------------------------
// Kernel 3: fused output via f32 WMMA outer product, channels-in-M mapping.
// One wave per (graph, 16-node tile):
//   A(16x4): row m = ( Wl[c0+m], constp[t,c0+m], 0, 0 )      (channels)
//   B(4x16): col n = ( s_n, 1, 0, 0 )^T                      (nodes)
//   D[m][n] = Wl_m * s_n + (bl+bias+pe)_m
// D layout: lane L holds node (L&15), channels c0 + (L>>4)*8 + {0..7} in 8
// consecutive VGPRs -> two aligned b128 stores per lane per channel tile.
// ---------------------------------------------------------------------------
__global__ void output_wmma_kernel(const float* __restrict__ Wl,
                                   const float* __restrict__ constp,
                                   const float* __restrict__ den,
                                   const float* __restrict__ snum,
                                   float* __restrict__ out) {
  const int TILES = (N_ + 15) / 16;  // 63
  int lane = threadIdx.x & 31;
  int wave = (blockIdx.x * blockDim.x + threadIdx.x) >> 5;
  if (wave >= G_ * TILES) return;    // wave-uniform

  int g     = wave / TILES;
  int tile  = wave - g * TILES;
  int t     = g % T_;
  int node0 = tile * 16;

  // B fragment (built once, reused across all 6 WMMAs):
  // lanes 0-15 carry K=0 (s) and K=1 (1.0); lanes 16-31 carry K=2,3 (zero)
  int  nodeL  = node0 + (lane & 15);
  bool nvalid = nodeL < N_;
  v2f bfrag = {0.0f, 0.0f};
  if (lane < 16) {
    float s = 0.0f;
    if (nvalid) {
      int gi = g * N_ + nodeL;
      s = snum[gi] / den[gi];   // den > 0 always (self loops)
    }
    bfrag[0] = s;
    bfrag[1] = 1.0f;
  }

  int half = lane >> 4;                         // channel sub-block (0 or 1)
  size_t gn = (size_t)g * N_ + (nvalid ? nodeL : 0);
  float* orow = out + gn * C_ + half * 8;       // 16B aligned (384|64|32 B terms)

#pragma unroll
  for (int ct = 0; ct < C_ / 16; ++ct) {
    int c0 = ct * 16;
    // A fragment: lanes 0-15 carry K=0 (Wl) and K=1 (const); lanes 16-31 zero
    v2f afrag = {0.0f, 0.0f};
    if (lane < 16) {
      afrag[0] = Wl[c0 + lane];
      afrag[1] = constp[t * C_ + c0 + lane];
    }
    v8f acc = {};
    acc = __builtin_amdgcn_wmma_f32_16x16x4_f32(
        /*neg_a=*/false, afrag, /*neg_b=*/false, bfrag,
        /*c_mod=*/(short)0, acc, /*reuse_a=*/false, /*reuse_b=*/false);

    if (nvalid) {
      float4* p = (float4*)(orow + c0);
      float4 lo, hi;
      lo.x = acc[0]; lo.y = acc[1]; lo.z = acc[2]; lo.w = acc[3];
      hi.x = acc[4]; hi.y = acc[5]; hi.z = acc[6]; hi.w = acc[7];
      p[0] = lo;     // channels c0 + half*8 + 0..3
      p[1] = hi;     // channels c0 + half*8 + 4..7
    }
  }
}

// ---------------------------------------------------------------------------
// Launch
// ---------------------------------------------------------------------------
extern "C" void kernel_launch(void* const* d_in, const int* in_sizes, int n_in,
                              void* d_out, int out_size, void* d_ws, size_t ws_size,
                              hipStream_t stream) {
  (void)in_sizes; (void)n_in; (void)out_size; (void)ws_size;
  const float* x    = (const float*)d_in[0];
  const int*   ei   = (const int*)d_in[1];
  const float* Wl   = (const float*)d_in[2];
  const float* bl   = (const float*)d_in[3];
  const float* Wr   = (const float*)d_in[4];
  const float* br   = (const float*)d_in[5];
  const float* att  = (const float*)d_in[6];
  const float* bias = (const float*)d_in[7];
  float* out = (float*)d_out;
  float* ws  = (float*)d_ws;

  const int GN = G_ * N_;
  float* m      = ws;               // [G*N]
  float* den    = ws + GN;          // [G*N]
  float* snum   = ws + 2 * GN;      // [G*N]
  float* constp = ws + 3 * GN;      // [T*C]
  float* tbl4   = constp + T_ * C_; // [4*C] (16B-aligned offset)

  {
    int n = GN;  // covers T*C and C ranges too
    init_kernel<<<(n + 255) / 256, 256, 0, stream>>>(Wl, bl, Wr, br, att, bias,
                                                     m, den, snum, constp, tbl4);
  }
  {
    int n = G_ * ET_;  // 2,880,000 edge-instances
    logits_max_kernel<<<(n + 255) / 256, 256, 0, stream>>>(x, ei, tbl4, m);
    softmax_accum_kernel<<<(n + 255) / 256, 256, 0, stream>>>(x, ei, tbl4, m, den, snum);
  }
  {
    int waves  = G_ * ((N_ + 15) / 16);     // 20160 waves
    int blocks = (waves * 32 + 255) / 256;  // 8 waves per 256-thread block
    output_wmma_kernel<<<blocks, 256, 0, stream>>>(Wl, constp, den, snum, out);
  }
}